// DualStreamBlock_38465727103446
// MI455X (gfx1250) — compile-verified
//
#include <hip/hip_runtime.h>
#include <hip/hip_bf16.h>

typedef __attribute__((ext_vector_type(2))) float v2f;
typedef __attribute__((ext_vector_type(8))) float v8f;

#define LSEQ  4096
#define LBITS 12

__device__ __forceinline__ float fsilu(float x)     { return x / (1.f + __expf(-x)); }
__device__ __forceinline__ float fsoftplus(float x) { return (x > 20.f) ? x : __logf(1.f + __expf(x)); }

// ---------------------------------------------------------------------------
// Generic WMMA-f32 GEMM:  out = epilogue( A @ W )
//   A is M x K (gathered per AMODE), W is K x N row-major.
// AMODE: 0 = row-major A1 (leading dim g.lda)
//        1 = A1 stored (B, K, L) i.e. transposed read; optional per-(b,k)
//            additive anchor A2 (may be null)
//        2 = concat [ A1 (M x K1) | A2 (M x (K-K1)) ], both row-major
// EPI:   0 = row-major store
//        1 = row-major store + bias
//        2 = row-major store + bias + SiLU
//        3 = transposed store into (B, N, L) + residual add from g.resid
//
// Tile staging uses GLOBAL_LOAD_ASYNC_TO_LDS_B128 (ASYNCcnt) whenever the
// tile is full and 16B-aligned; ragged K/N tiles fall back to guarded VALU
// fills (which also provide the required zero padding).
// ---------------------------------------------------------------------------
struct GemmArgs {
  const float* A1; const float* A2; const float* W; const float* bias; const float* resid;
  float* out;
  int M, N, K, K1, lda;
};

template<int AMODE, int EPI>
__global__ void gemm_wmma_f32(GemmArgs g) {
  __shared__ float As[64][20];   // row stride 80B: 16B-aligned quads, conflict-free cols
  __shared__ float Bs[16][64];

  const int tid    = threadIdx.x;          // 128 threads = 4 waves (wave32)
  const int lane   = tid & 31;
  const int wave   = tid >> 5;
  const int lanelo = lane & 15;
  const int lanehi = lane >> 4;
  const int M0 = blockIdx.x * 64;
  const int N0 = blockIdx.y * 64;

  const v8f vzero = {0.f,0.f,0.f,0.f,0.f,0.f,0.f,0.f};
  v8f acc[4];
  acc[0] = vzero; acc[1] = vzero; acc[2] = vzero; acc[3] = vzero;

  const int nkt = (g.K + 15) >> 4;
  for (int kt = 0; kt < nkt; ++kt) {
    const int k0 = kt << 4;
    const bool fullK = (k0 + 16) <= g.K;          // uniform across block
    const bool fullN = (N0 + 64) <= g.N;

    // ---- stage A tile ----
    const bool asyncA = (AMODE != 1) && fullK && (AMODE == 2 || (g.lda & 3) == 0);
    if (asyncA) {
      // 2 x b128 per thread: async DMA into LDS, no VGPR data movement
      #pragma unroll
      for (int q = 0; q < 2; ++q) {
        int e = (tid + q * 128) * 4;              // 0..1020
        int m = e >> 4, k = e & 15;
        int gm = M0 + m, gk = k0 + k;
        const float* gp;
        if (AMODE == 0) gp = g.A1 + (size_t)gm * g.lda + gk;
        else            gp = (gk < g.K1) ? (g.A1 + (size_t)gm * g.K1 + gk)
                                         : (g.A2 + (size_t)gm * (g.K - g.K1) + (gk - g.K1));
        unsigned la = (unsigned)(size_t)&As[m][k];   // addr[31:0] == LDS byte offset
        asm volatile("global_load_async_to_lds_b128 %0, %1, off"
                     :: "v"(la), "v"(gp) : "memory");
      }
    } else if (AMODE == 1) {
      // consecutive threads -> consecutive m -> consecutive t : coalesced
      for (int e = tid; e < 64 * 16; e += 128) {
        int m = e & 63, k = e >> 6;
        int gk = k0 + k; float v = 0.f;
        if (gk < g.K) {
          int gm = M0 + m, b = gm >> LBITS, t = gm & (LSEQ - 1);
          v = g.A1[((size_t)b * g.K + gk) * LSEQ + t];
          if (g.A2) v += g.A2[b * g.K + gk];
        }
        As[m][k] = v;
      }
    } else {
      for (int e = tid; e < 64 * 16; e += 128) {
        int m = e >> 4, k = e & 15;
        int gk = k0 + k; float v = 0.f;
        int gm = M0 + m;
        if (gk < g.K) {
          if (AMODE == 0) v = g.A1[(size_t)gm * g.lda + gk];
          else            v = (gk < g.K1) ? g.A1[(size_t)gm * g.K1 + gk]
                                          : g.A2[(size_t)gm * (g.K - g.K1) + (gk - g.K1)];
        }
        As[m][k] = v;
      }
    }

    // ---- stage W tile ----
    if (fullK && fullN) {
      #pragma unroll
      for (int q = 0; q < 2; ++q) {
        int e = (tid + q * 128) * 4;
        int k = e >> 6, n = e & 63;
        const float* gp = g.W + (size_t)(k0 + k) * g.N + N0 + n;
        unsigned la = (unsigned)(size_t)&Bs[k][n];
        asm volatile("global_load_async_to_lds_b128 %0, %1, off"
                     :: "v"(la), "v"(gp) : "memory");
      }
    } else {
      for (int e = tid; e < 16 * 64; e += 128) {
        int k = e >> 6, n = e & 63;
        int gk = k0 + k, gn = N0 + n;
        Bs[k][n] = (gk < g.K && gn < g.N) ? g.W[(size_t)gk * g.N + gn] : 0.f;
      }
    }

    // ---- prefetch next W k-rows into cache (global_prefetch_b8) ----
    if (kt + 1 < nkt) {
      int pk = k0 + 16 + (tid & 15);
      int pn = N0 + ((tid >> 4) << 3);
      if (pk < g.K && pn < g.N)
        __builtin_prefetch(&g.W[(size_t)pk * g.N + pn], 0, 1);
    }

    asm volatile("s_wait_asynccnt 0x0" ::: "memory");   // async LDS writes landed
    __syncthreads();

    // ---- 4 WMMA k-steps of 4 each; A frag shared across the 4 n-tiles ----
    #pragma unroll
    for (int kk = 0; kk < 16; kk += 4) {
      const int kb = kk + 2 * lanehi;                 // ISA: A/B k = 2*(lane/16) + vgpr
      v2f a;
      a.x = As[wave * 16 + lanelo][kb];
      a.y = As[wave * 16 + lanelo][kb + 1];
      #pragma unroll
      for (int j = 0; j < 4; ++j) {
        v2f b;
        b.x = Bs[kb][j * 16 + lanelo];
        b.y = Bs[kb + 1][j * 16 + lanelo];
        acc[j] = __builtin_amdgcn_wmma_f32_16x16x4_f32(
            false, a, false, b, (short)0, acc[j], false, false);
      }
    }
    __syncthreads();
  }

  // ---- epilogue; C/D layout: m = vgpr + 8*(lane/16), n = lane%16 ----
  #pragma unroll
  for (int j = 0; j < 4; ++j) {
    const int n = N0 + j * 16 + lanelo;
    if (n < g.N) {
      float bv = (EPI == 1 || EPI == 2) ? g.bias[n] : 0.f;
      #pragma unroll
      for (int rr = 0; rr < 8; ++rr) {
        int m = M0 + wave * 16 + rr + 8 * lanehi;
        float v = acc[j][rr] + bv;
        if (EPI == 2) v = fsilu(v);
        if (EPI == 3) {
          int b = m >> LBITS, t = m & (LSEQ - 1);
          size_t o = ((size_t)b * g.N + n) * LSEQ + t;
          g.out[o] = v + g.resid[o];
        } else {
          g.out[(size_t)m * g.N + n] = v;
        }
      }
    }
  }
}

// ---------------------------------------------------------------------------
// Causal depthwise conv1d (k=4, left pad 3) + SiLU.
// ---------------------------------------------------------------------------
__global__ void dwconv_silu(const float* __restrict__ xin, int lda,
                            const float* __restrict__ w, const float* __restrict__ cb,
                            float* __restrict__ xc, int dchan, int total) {
  int idx = blockIdx.x * blockDim.x + threadIdx.x;
  if (idx >= total) return;
  int d  = idx % dchan;
  int bt = idx / dchan;
  int t  = bt & (LSEQ - 1);
  float acc = cb[d];
  #pragma unroll
  for (int j = 0; j < 4; ++j) {
    int tt = t - 3 + j;
    if (tt >= 0) acc = fmaf(w[d * 4 + j], xin[(size_t)(bt - 3 + j) * lda + d], acc);
  }
  xc[(size_t)bt * dchan + d] = fsilu(acc);
}

// ---------------------------------------------------------------------------
// Row LayerNorm over channels of X (B, C, L); output kept in (B, C, L) layout
// so the consuming GEMM uses the coalesced transposed-read path (AMODE 1).
// ---------------------------------------------------------------------------
__global__ void ln_rows(const float* __restrict__ X, const float* __restrict__ gam,
                        const float* __restrict__ bet, float* __restrict__ out, int C) {
  int row = blockIdx.x * blockDim.x + threadIdx.x;   // b*L + t
  int b = row >> LBITS, t = row & (LSEQ - 1);
  const float* xp = X + (size_t)b * C * LSEQ + t;
  float s = 0.f, ss = 0.f;
  for (int c = 0; c < C; ++c) {
    float v = xp[(size_t)c * LSEQ];
    s += v; ss += v * v;
  }
  float mean = s / C;
  float var  = ss / C - mean * mean;
  float rs   = rsqrtf(var + 1e-5f);
  float* op = out + (size_t)b * C * LSEQ + t;
  for (int c = 0; c < C; ++c) {
    float v = xp[(size_t)c * LSEQ];
    op[(size_t)c * LSEQ] = (v - mean) * rs * gam[c] + bet[c];
  }
}

// ---------------------------------------------------------------------------
// Chunked parallel selective scan: L=4096 -> 64 chunks of 64.
// Lane n (0..15) owns one of the N=16 states.
// ---------------------------------------------------------------------------
struct ScanArgs {
  const float* u; const float* dtf; const float* xdbl;
  const float* dtb; const float* Alog; const float* Dp; const float* z;
  float* y; float* P; float* Q; float* H;
  int dchan, ncol, roff, zlda, zoff;
};

__global__ void scan_chunk1(ScanArgs s) {    // per-chunk (P = prod a, q = affine const)
  int g  = blockIdx.x * blockDim.y + threadIdx.y;   // ch*64 + chunk
  int ch = g >> 6, ck = g & 63;
  int b  = ch / s.dchan, d = ch - b * s.dchan;
  int n  = threadIdx.x;
  float A  = -__expf(s.Alog[d * 16 + n]);
  float db = s.dtb[d];
  float P = 1.f, q = 0.f;
  size_t row = ((size_t)b << LBITS) + (size_t)ck * 64;
  for (int i = 0; i < 64; ++i, ++row) {
    float dv = fsoftplus(s.dtf[row * s.dchan + d] + db);
    float a  = __expf(dv * A);
    float bu = dv * s.u[row * s.dchan + d] * s.xdbl[row * s.ncol + s.roff + n];
    q = fmaf(a, q, bu);
    P *= a;
  }
  size_t ci = ((size_t)g << 4) + n;
  s.P[ci] = P; s.Q[ci] = q;
}

__global__ void scan_chunk2(ScanArgs s) {    // tiny 64-step carry scan per channel
  int ch = blockIdx.x, n = threadIdx.x;
  float h = 0.f;
  size_t base = ((size_t)ch << 6);
  for (int ck = 0; ck < 64; ++ck) {
    size_t ci = ((base + ck) << 4) + n;
    s.H[ci] = h;
    h = fmaf(s.P[ci], h, s.Q[ci]);
  }
}

__global__ void scan_chunk3(ScanArgs s) {    // replay with correct h0, emit gated y
  int g  = blockIdx.x * blockDim.y + threadIdx.y;
  int ch = g >> 6, ck = g & 63;
  int b  = ch / s.dchan, d = ch - b * s.dchan;
  int n  = threadIdx.x;
  float A  = -__expf(s.Alog[d * 16 + n]);
  float db = s.dtb[d];
  float Dv = s.Dp[d];
  float h  = s.H[((size_t)g << 4) + n];
  size_t row = ((size_t)b << LBITS) + (size_t)ck * 64;
  for (int i = 0; i < 64; ++i, ++row) {
    float dv = fsoftplus(s.dtf[row * s.dchan + d] + db);
    float a  = __expf(dv * A);
    float uu = s.u[row * s.dchan + d];
    float bu = dv * uu * s.xdbl[row * s.ncol + s.roff + n];
    h = fmaf(a, h, bu);
    float p = h * s.xdbl[row * s.ncol + s.roff + 16 + n];
    p += __shfl_xor(p, 1, 32);     // 16-lane tree reduction (groups aligned to 16)
    p += __shfl_xor(p, 2, 32);
    p += __shfl_xor(p, 4, 32);
    p += __shfl_xor(p, 8, 32);
    if (n == 0) {
      float yv = p + Dv * uu;
      float zv = s.z[row * s.zlda + s.zoff + d];
      s.y[row * s.dchan + d] = yv * fsilu(zv);
    }
  }
}

// ---------------------------------------------------------------------------
extern "C" void kernel_launch(void* const* d_in, const int* in_sizes, int n_in,
                              void* d_out, int out_size, void* d_ws, size_t ws_size,
                              hipStream_t stream) {
  (void)in_sizes; (void)n_in; (void)out_size; (void)ws_size;
  const float* m_ll    = (const float*)d_in[0];
  const float* m_high  = (const float*)d_in[1];
  const float* zanc    = (const float*)d_in[2];
  const float* Kin     = (const float*)d_in[3];
  const float* Qin     = (const float*)d_in[4];
  const float* s_in_w  = (const float*)d_in[5];
  const float* s_conv_w= (const float*)d_in[6];
  const float* s_conv_b= (const float*)d_in[7];
  const float* s_x_w   = (const float*)d_in[8];
  const float* s_dt_w  = (const float*)d_in[9];
  const float* s_dt_b  = (const float*)d_in[10];
  const float* s_Alog  = (const float*)d_in[11];
  const float* s_D     = (const float*)d_in[12];
  const float* s_out_w = (const float*)d_in[13];
  const float* t_in_w  = (const float*)d_in[14];
  const float* t_conv_w= (const float*)d_in[15];
  const float* t_conv_b= (const float*)d_in[16];
  const float* t_kln_g = (const float*)d_in[17];
  const float* t_kln_b = (const float*)d_in[18];
  const float* t_k_w   = (const float*)d_in[19];
  const float* t_k_b   = (const float*)d_in[20];
  const float* t_qln_g = (const float*)d_in[21];
  const float* t_qln_b = (const float*)d_in[22];
  const float* t_q_w   = (const float*)d_in[23];
  const float* t_q_b   = (const float*)d_in[24];
  const float* t_dtbc_w= (const float*)d_in[25];
  const float* t_dt_w  = (const float*)d_in[26];
  const float* t_dt_b  = (const float*)d_in[27];
  const float* t_gate_w= (const float*)d_in[28];
  const float* t_gate_b= (const float*)d_in[29];
  const float* t_Alog  = (const float*)d_in[30];
  const float* t_D     = (const float*)d_in[31];
  const float* t_out_w = (const float*)d_in[32];

  float* o1 = (float*)d_out;                       // (B, 96, L)
  float* o2 = o1 + (size_t)4 * 96 * LSEQ;          // (B, 288, L)

  // Workspace regions (reused across phases; nothing live overlaps)
  float* ws = (float*)d_ws;
  const size_t SZ = (size_t)16384 * 576;
  float* R0  = ws;                                  // xin (S:384 / T:576 cols)
  float* R1  = R0 + SZ;                             // xc
  float* R2  = R1 + SZ;                             // lnK -> lnQ -> gate
  float* R3  = R2 + SZ;                             // Kp -> dtfull
  float* R4  = R3 + SZ;                             // Qp -> y
  float* R5  = R4 + SZ;                             // xdbl (<= 16384*64)
  float* CP  = R5 + (size_t)16384 * 64;             // chunk carries
  float* CQ  = CP + (size_t)2304 * 1024;
  float* CHb = CQ + (size_t)2304 * 1024;

  const int M = 16384;
  dim3 blk(128);

  // ================= S-stream (DIM=96, di=192) =================
  { GemmArgs a{}; a.A1 = m_ll; a.A2 = zanc; a.W = s_in_w; a.out = R0;
    a.M = M; a.N = 384; a.K = 96;
    gemm_wmma_f32<1,0><<<dim3(256, 6), blk, 0, stream>>>(a); }
  dwconv_silu<<<(M*192 + 255)/256, 256, 0, stream>>>(R0, 384, s_conv_w, s_conv_b, R1, 192, M*192);
  { GemmArgs a{}; a.A1 = R1; a.lda = 192; a.W = s_x_w; a.out = R5;
    a.M = M; a.N = 38; a.K = 192;
    gemm_wmma_f32<0,0><<<dim3(256, 1), blk, 0, stream>>>(a); }
  { GemmArgs a{}; a.A1 = R5; a.lda = 38; a.W = s_dt_w; a.out = R3;
    a.M = M; a.N = 192; a.K = 6;
    gemm_wmma_f32<0,0><<<dim3(256, 3), blk, 0, stream>>>(a); }
  { ScanArgs sc{R1, R3, R5, s_dt_b, s_Alog, s_D, R0, R4, CP, CQ, CHb, 192, 38, 6, 384, 192};
    scan_chunk1<<<768*64/4, dim3(16, 4), 0, stream>>>(sc);
    scan_chunk2<<<768, 16, 0, stream>>>(sc);
    scan_chunk3<<<768*64/4, dim3(16, 4), 0, stream>>>(sc); }
  { GemmArgs a{}; a.A1 = R4; a.lda = 192; a.W = s_out_w; a.resid = m_ll; a.out = o1;
    a.M = M; a.N = 96; a.K = 192;
    gemm_wmma_f32<0,3><<<dim3(256, 2), blk, 0, stream>>>(a); }

  // ================= T-stream (DIM3=288, di3=576) =================
  { GemmArgs a{}; a.A1 = m_high; a.A2 = nullptr; a.W = t_in_w; a.out = R0;
    a.M = M; a.N = 576; a.K = 288;
    gemm_wmma_f32<1,0><<<dim3(256, 9), blk, 0, stream>>>(a); }
  dwconv_silu<<<(M*576 + 255)/256, 256, 0, stream>>>(R0, 576, t_conv_w, t_conv_b, R1, 576, M*576);
  ln_rows<<<64, 256, 0, stream>>>(Kin, t_kln_g, t_kln_b, R2, 288);
  { GemmArgs a{}; a.A1 = R2; a.A2 = nullptr; a.W = t_k_w; a.bias = t_k_b; a.out = R3;
    a.M = M; a.N = 576; a.K = 288;
    gemm_wmma_f32<1,2><<<dim3(256, 9), blk, 0, stream>>>(a); }
  ln_rows<<<64, 256, 0, stream>>>(Qin, t_qln_g, t_qln_b, R2, 288);
  { GemmArgs a{}; a.A1 = R2; a.A2 = nullptr; a.W = t_q_w; a.bias = t_q_b; a.out = R4;
    a.M = M; a.N = 576; a.K = 288;
    gemm_wmma_f32<1,2><<<dim3(256, 9), blk, 0, stream>>>(a); }
  { GemmArgs a{}; a.A1 = R1; a.A2 = R3; a.K1 = 576; a.W = t_dtbc_w; a.out = R5;
    a.M = M; a.N = 50; a.K = 1152;
    gemm_wmma_f32<2,0><<<dim3(256, 1), blk, 0, stream>>>(a); }
  { GemmArgs a{}; a.A1 = R1; a.A2 = R4; a.K1 = 576; a.W = t_gate_w; a.bias = t_gate_b; a.out = R2;
    a.M = M; a.N = 576; a.K = 1152;
    gemm_wmma_f32<2,1><<<dim3(256, 9), blk, 0, stream>>>(a); }
  { GemmArgs a{}; a.A1 = R5; a.lda = 50; a.W = t_dt_w; a.out = R3;
    a.M = M; a.N = 576; a.K = 18;
    gemm_wmma_f32<0,0><<<dim3(256, 9), blk, 0, stream>>>(a); }
  { ScanArgs sc{R1, R3, R5, t_dt_b, t_Alog, t_D, R2, R4, CP, CQ, CHb, 576, 50, 18, 576, 0};
    scan_chunk1<<<2304*64/4, dim3(16, 4), 0, stream>>>(sc);
    scan_chunk2<<<2304, 16, 0, stream>>>(sc);
    scan_chunk3<<<2304*64/4, dim3(16, 4), 0, stream>>>(sc); }
  { GemmArgs a{}; a.A1 = R4; a.lda = 576; a.W = t_out_w; a.resid = m_high; a.out = o2;
    a.M = M; a.N = 288; a.K = 576;
    gemm_wmma_f32<0,3><<<dim3(256, 5), blk, 0, stream>>>(a); }
}